// NeighRoutingGnnCls2Scores_65025804861639
// MI455X (gfx1250) — compile-verified
//
#include <hip/hip_runtime.h>
#include <hip/hip_bf16.h>
#include <math.h>

#define NR     49999   // rows of x / iv / cls_tab (item_emb[1:])
#define HD     128
#define MNB    16
#define BATCH  512
#define SEQL   50
#define KFACT  10
#define GRUB   1024    // combined GRU batch (session 0..511, cls 512..1023)
#define XROWS  51200   // GRUB * SEQL
#define G3     384     // 3*HD gates
#define SESSN  25600   // BATCH * SEQL

typedef __attribute__((ext_vector_type(2))) float v2f;
typedef __attribute__((ext_vector_type(8))) float v8f;

__device__ __forceinline__ float wave_sum(float v) {
#pragma unroll
  for (int off = 16; off > 0; off >>= 1) v += __shfl_xor(v, off, 32);
  return v;
}
__device__ __forceinline__ float sigmoidf(float x) { return 1.0f / (1.0f + __expf(-x)); }

// LayerNorm of a 128-wide row held as 4 floats/lane across one wave32.
__device__ __forceinline__ float4 ln4(float4 v, const float* __restrict__ g,
                                      const float* __restrict__ b, int lane) {
  float mu = wave_sum(v.x + v.y + v.z + v.w) * (1.0f / HD);
  float dx = v.x - mu, dy = v.y - mu, dz = v.z - mu, dw = v.w - mu;
  float var = wave_sum(dx * dx + dy * dy + dz * dz + dw * dw) * (1.0f / HD);
  float inv = rsqrtf(var + 1e-5f);
  float4 gg = *reinterpret_cast<const float4*>(g + lane * 4);
  float4 bb = *reinterpret_cast<const float4*>(b + lane * 4);
  return make_float4(dx * inv * gg.x + bb.x, dy * inv * gg.y + bb.y,
                     dz * inv * gg.z + bb.z, dw * inv * gg.w + bb.w);
}

// ---------------- row kernels (one wave32 per 128-float row) ----------------

__global__ void k_normalize(const float* __restrict__ in, float* __restrict__ out, int nrows) {
  int wid = (blockIdx.x * blockDim.x + threadIdx.x) >> 5;
  int lane = threadIdx.x & 31;
  if (wid >= nrows) return;
  float4 v = *reinterpret_cast<const float4*>(in + (size_t)wid * HD + lane * 4);
  float ss = wave_sum(v.x * v.x + v.y * v.y + v.z * v.z + v.w * v.w);
  float inv = 1.0f / fmaxf(sqrtf(ss), 1e-12f);
  *reinterpret_cast<float4*>(out + (size_t)wid * HD + lane * 4) =
      make_float4(v.x * inv, v.y * inv, v.z * inv, v.w * inv);
}

__global__ void k_routing(const float* __restrict__ xn, const int* __restrict__ adj,
                          float* __restrict__ out, int nrows) {
  int wid = (blockIdx.x * blockDim.x + threadIdx.x) >> 5;
  int lane = threadIdx.x & 31;
  if (wid >= nrows) return;
  const int* nb = adj + (size_t)(wid + 1) * MNB;  // adj_items[1:]
  float4 x = *reinterpret_cast<const float4*>(xn + (size_t)wid * HD + lane * 4);
  float4 z[MNB];
  float ux = 0.f, uy = 0.f, uz = 0.f, uw = 0.f;
#pragma unroll
  for (int m = 0; m < MNB; ++m) {
    int idx = nb[m] - 1;
    z[m] = *reinterpret_cast<const float4*>(xn + (size_t)idx * HD + lane * 4);
    ux += z[m].x; uy += z[m].y; uz += z[m].z; uw += z[m].w;
  }
  const float invm = 1.0f / (float)MNB;
  ux = ux * invm + x.x; uy = uy * invm + x.y; uz = uz * invm + x.z; uw = uw * invm + x.w;
#pragma unroll 1
  for (int it = 0; it < 3; ++it) {
    float sq = wave_sum(ux * ux + uy * uy + uz * uz + uw * uw);
    float scale = (sq / (sq + 1.0f)) / fmaxf(sqrtf(sq), 1e-12f);
    ux *= scale; uy *= scale; uz *= scale; uw *= scale;
    float lg[MNB];
#pragma unroll
    for (int m = 0; m < MNB; ++m)
      lg[m] = z[m].x * ux + z[m].y * uy + z[m].z * uz + z[m].w * uw;
#pragma unroll
    for (int off = 16; off > 0; off >>= 1)
#pragma unroll
      for (int m = 0; m < MNB; ++m) lg[m] += __shfl_xor(lg[m], off, 32);
    float mx = lg[0];
#pragma unroll
    for (int m = 1; m < MNB; ++m) mx = fmaxf(mx, lg[m]);
    float s = 0.f;
#pragma unroll
    for (int m = 0; m < MNB; ++m) { lg[m] = __expf(lg[m] - mx); s += lg[m]; }
    float invs = 1.0f / s;
    float nx = x.x, ny = x.y, nz = x.z, nw = x.w;
#pragma unroll
    for (int m = 0; m < MNB; ++m) {
      float p = lg[m] * invs;
      nx += z[m].x * p; ny += z[m].y * p; nz += z[m].z * p; nw += z[m].w * p;
    }
    ux = nx; uy = ny; uz = nz; uw = nw;
  }
  *reinterpret_cast<float4*>(out + (size_t)wid * HD + lane * 4) = make_float4(ux, uy, uz, uw);
}

__global__ void k_sum_ln(const float* __restrict__ a, const float* __restrict__ b,
                         const float* __restrict__ c, const float* __restrict__ g,
                         const float* __restrict__ be, float* __restrict__ out, int nrows) {
  int wid = (blockIdx.x * blockDim.x + threadIdx.x) >> 5;
  int lane = threadIdx.x & 31;
  if (wid >= nrows) return;
  size_t off = (size_t)wid * HD + lane * 4;
  float4 va = *reinterpret_cast<const float4*>(a + off);
  float4 vb = *reinterpret_cast<const float4*>(b + off);
  float4 vc = *reinterpret_cast<const float4*>(c + off);
  float4 v = make_float4(va.x + vb.x + vc.x, va.y + vb.y + vc.y,
                         va.z + vb.z + vc.z, va.w + vb.w + vc.w);
  *reinterpret_cast<float4*>(out + off) = ln4(v, g, be, lane);
}

// out[r] = LN3( prob_emb[prow] @ W_cls^T ); prow = sess ? sess[r] : r+1
__global__ void k_cls_rows(const float* __restrict__ prob, const float* __restrict__ Wcls,
                           const float* __restrict__ g, const float* __restrict__ be,
                           const int* __restrict__ sess, float* __restrict__ out, int nrows) {
  int wid = (blockIdx.x * blockDim.x + threadIdx.x) >> 5;
  int lane = threadIdx.x & 31;
  if (wid >= nrows) return;
  int prow = sess ? sess[wid] : wid + 1;
  const float* pe = prob + (size_t)prow * KFACT;
  float p[KFACT];
#pragma unroll
  for (int j = 0; j < KFACT; ++j) p[j] = pe[j];
  float acc[4];
#pragma unroll
  for (int d = 0; d < 4; ++d) {
    const float* w = Wcls + (size_t)(lane * 4 + d) * KFACT;
    float s = 0.f;
#pragma unroll
    for (int j = 0; j < KFACT; ++j) s += p[j] * w[j];
    acc[d] = s;
  }
  float4 v = make_float4(acc[0], acc[1], acc[2], acc[3]);
  *reinterpret_cast<float4*>(out + (size_t)wid * HD + lane * 4) = ln4(v, g, be, lane);
}

__global__ void k_gather_emb(const float* __restrict__ iv, const int* __restrict__ sess,
                             float* __restrict__ X, int nrows) {
  int wid = (blockIdx.x * blockDim.x + threadIdx.x) >> 5;
  int lane = threadIdx.x & 31;
  if (wid >= nrows) return;
  int idx = sess[wid];
  float4 v = make_float4(0.f, 0.f, 0.f, 0.f);
  if (idx > 0) v = *reinterpret_cast<const float4*>(iv + (size_t)(idx - 1) * HD + lane * 4);
  *reinterpret_cast<float4*>(X + (size_t)wid * HD + lane * 4) = v;
}

__global__ void k_zero(float* __restrict__ p, int n) {
  int i = blockIdx.x * blockDim.x + threadIdx.x;
  if (i < n) p[i] = 0.f;
}

// ---------------- WMMA fp32 GEMM: C[M x N] = A[M x K] * B[N x K]^T ----------
// All dims multiples of 16. One wave per 16x16 C tile. V_WMMA_F32_16X16X4_F32.

__global__ void k_gemm_nt(const float* __restrict__ A, const float* __restrict__ B,
                          float* __restrict__ C, int M, int N, int K) {
  int wid = (blockIdx.x * blockDim.x + threadIdx.x) >> 5;
  int ntn = N >> 4;
  if (wid >= (M >> 4) * ntn) return;
  int mt = wid / ntn, nt = wid % ntn;
  int lane = threadIdx.x & 31, half = lane >> 4, l16 = lane & 15;
  const float* Ap = A + (size_t)(mt * 16 + l16) * K + half * 2;
  const float* Bp = B + (size_t)(nt * 16 + l16) * K + half * 2;
  v8f acc = {};
#pragma unroll 4
  for (int k = 0; k < K; k += 4) {
    v2f a = *reinterpret_cast<const v2f*>(Ap + k);
    v2f b = *reinterpret_cast<const v2f*>(Bp + k);
    acc = __builtin_amdgcn_wmma_f32_16x16x4_f32(false, a, false, b, (short)0, acc,
                                                false, false);
  }
  float* Cp = C + (size_t)nt * 16 + l16;
#pragma unroll
  for (int g = 0; g < 8; ++g)
    Cp[(size_t)(mt * 16 + g + half * 8) * N] = acc[g];
}

// ---------------- GRU gate fusion (one element per thread) ------------------

__global__ void k_gru_gate(const float* __restrict__ gi, const float* __restrict__ gh,
                           const float* __restrict__ bih, const float* __restrict__ bhh,
                           const float* __restrict__ Hprev, float* __restrict__ Hout, int t) {
  int gid = blockIdx.x * blockDim.x + threadIdx.x;
  if (gid >= GRUB * HD) return;
  int i = gid >> 7, h = gid & 127;
  int gir = (i < BATCH) ? (i * SEQL + t) : (SESSN + (i - BATCH) * SEQL + t);
  const float* gip = gi + (size_t)gir * G3;
  const float* ghp = gh + (size_t)i * G3;
  float ir = gip[h] + bih[h];
  float iz = gip[HD + h] + bih[HD + h];
  float in_ = gip[2 * HD + h] + bih[2 * HD + h];
  float hr = ghp[h] + bhh[h];
  float hz = ghp[HD + h] + bhh[HD + h];
  float hn = ghp[2 * HD + h] + bhh[2 * HD + h];
  float r = sigmoidf(ir + hr);
  float zg = sigmoidf(iz + hz);
  float ng = tanhf(in_ + r * hn);
  Hout[gid] = (1.0f - zg) * ng + zg * Hprev[gid];
}

__global__ void k_select_ht(const float* __restrict__ Hbuf, const int* __restrict__ lengths,
                            const float* __restrict__ g2, const float* __restrict__ b2,
                            const float* __restrict__ g4, const float* __restrict__ b4,
                            float* __restrict__ ht, float* __restrict__ htc) {
  int wid = (blockIdx.x * blockDim.x + threadIdx.x) >> 5;
  int lane = threadIdx.x & 31;
  if (wid >= BATCH) return;
  int t = lengths[wid] - 1;
  const float* base = Hbuf + (size_t)t * GRUB * HD;
  float4 v = *reinterpret_cast<const float4*>(base + (size_t)wid * HD + lane * 4);
  *reinterpret_cast<float4*>(ht + (size_t)wid * HD + lane * 4) = ln4(v, g2, b2, lane);
  float4 w = *reinterpret_cast<const float4*>(base + (size_t)(BATCH + wid) * HD + lane * 4);
  *reinterpret_cast<float4*>(htc + (size_t)wid * HD + lane * 4) = ln4(w, g4, b4, lane);
}

// ---------------- dual WMMA scores + combine (NT stores, 307 MB stream) -----

__global__ void k_scores(const float* __restrict__ ht, const float* __restrict__ htc,
                         const float* __restrict__ iv, const float* __restrict__ ct,
                         const float* __restrict__ a1, const float* __restrict__ a2,
                         float* __restrict__ out) {
  const int NTN = (NR + 15) >> 4;  // 3125
  int wid = (blockIdx.x * blockDim.x + threadIdx.x) >> 5;
  if (wid >= (BATCH / 16) * NTN) return;
  int mt = wid / NTN, nt = wid % NTN;
  int lane = threadIdx.x & 31, half = lane >> 4, l16 = lane & 15;
  int brow = nt * 16 + l16;
  if (brow > NR - 1) brow = NR - 1;  // clamp B loads at the ragged edge
  const float* Ap1 = ht + (size_t)(mt * 16 + l16) * HD + half * 2;
  const float* Ap2 = htc + (size_t)(mt * 16 + l16) * HD + half * 2;
  const float* Bp1 = iv + (size_t)brow * HD + half * 2;
  const float* Bp2 = ct + (size_t)brow * HD + half * 2;
  v8f acc1 = {};
  v8f acc2 = {};
#pragma unroll 4
  for (int k = 0; k < HD; k += 4) {
    v2f aa = *reinterpret_cast<const v2f*>(Ap1 + k);
    v2f bb = *reinterpret_cast<const v2f*>(Bp1 + k);
    acc1 = __builtin_amdgcn_wmma_f32_16x16x4_f32(false, aa, false, bb, (short)0, acc1,
                                                 false, false);
    v2f cc = *reinterpret_cast<const v2f*>(Ap2 + k);
    v2f dd = *reinterpret_cast<const v2f*>(Bp2 + k);
    acc2 = __builtin_amdgcn_wmma_f32_16x16x4_f32(false, cc, false, dd, (short)0, acc2,
                                                 false, false);
  }
  float s1 = sigmoidf(a1[0]), s2 = sigmoidf(a2[0]);
  int ncol = nt * 16 + l16;
  if (ncol < NR) {
    const size_t SC = (size_t)BATCH * NR;
#pragma unroll
    for (int g = 0; g < 8; ++g) {
      size_t off = (size_t)(mt * 16 + g + half * 8) * NR + ncol;
      float v1 = acc1[g], v2 = acc2[g];
      __builtin_nontemporal_store(s1 * v1 + s2 * v2, out + off);
      __builtin_nontemporal_store(v1, out + SC + off);
      __builtin_nontemporal_store(v2, out + 2 * SC + off);
    }
  }
}

// ---------------------------------------------------------------------------

extern "C" void kernel_launch(void* const* d_in, const int* in_sizes, int n_in,
                              void* d_out, int out_size, void* d_ws, size_t ws_size,
                              hipStream_t stream) {
  (void)in_sizes; (void)n_in; (void)out_size; (void)ws_size;
  const float* item_emb = (const float*)d_in[0];
  const float* prob_emb = (const float*)d_in[1];
  const float* W_cls    = (const float*)d_in[2];
  const float* a1       = (const float*)d_in[3];
  const float* a2       = (const float*)d_in[4];
  const float* Wih      = (const float*)d_in[5];
  const float* Whh      = (const float*)d_in[6];
  const float* bih      = (const float*)d_in[7];
  const float* bhh      = (const float*)d_in[8];
  const float* ln1g = (const float*)d_in[9],  *ln1b = (const float*)d_in[10];
  const float* ln2g = (const float*)d_in[11], *ln2b = (const float*)d_in[12];
  const float* ln3g = (const float*)d_in[13], *ln3b = (const float*)d_in[14];
  const float* ln4g = (const float*)d_in[15], *ln4b = (const float*)d_in[16];
  // d_in[17]=mask_1, d_in[18]=mask_inf: unused by the math
  const int* inp_sess = (const int*)d_in[19];
  const int* lengths  = (const int*)d_in[20];
  const int* adj      = (const int*)d_in[21];
  float* out = (float*)d_out;

  float* ws = (float*)d_ws;
  size_t o = 0;
  float* xn   = ws + o; o += (size_t)NR * HD;
  float* out1 = ws + o; o += (size_t)NR * HD;
  float* out2 = ws + o; o += (size_t)NR * HD;
  float* iv   = ws + o; o += (size_t)NR * HD;
  float* ctab = ws + o; o += (size_t)NR * HD;
  float* X    = ws + o; o += (size_t)XROWS * HD;
  float* gi   = ws + o; o += (size_t)XROWS * G3;
  float* gh   = ws + o; o += (size_t)GRUB * G3;
  float* Hbuf = ws + o; o += (size_t)SEQL * GRUB * HD;
  float* h0   = ws + o; o += (size_t)GRUB * HD;
  float* ht   = ws + o; o += (size_t)BATCH * HD;
  float* htc  = ws + o; o += (size_t)BATCH * HD;

  const float* x = item_emb + HD;  // item_emb[1:]
  dim3 blk(256);
  int rowBlocks = (NR + 7) / 8;

  // --- 2-hop capsule routing ---
  k_normalize<<<rowBlocks, blk, 0, stream>>>(x, xn, NR);
  k_routing  <<<rowBlocks, blk, 0, stream>>>(xn, adj, out1, NR);
  k_normalize<<<rowBlocks, blk, 0, stream>>>(out1, xn, NR);
  k_routing  <<<rowBlocks, blk, 0, stream>>>(xn, adj, out2, NR);
  k_sum_ln   <<<rowBlocks, blk, 0, stream>>>(x, out1, out2, ln1g, ln1b, iv, NR);

  // --- class tables + GRU input assembly ---
  k_cls_rows<<<rowBlocks, blk, 0, stream>>>(prob_emb, W_cls, ln3g, ln3b, nullptr, ctab, NR);
  k_cls_rows<<<(SESSN + 7) / 8, blk, 0, stream>>>(prob_emb, W_cls, ln3g, ln3b, inp_sess,
                                                  X + (size_t)SESSN * HD, SESSN);
  k_gather_emb<<<(SESSN + 7) / 8, blk, 0, stream>>>(iv, inp_sess, X, SESSN);

  // --- gi = X @ Wih^T (51200 x 384 x 128) one-shot WMMA GEMM ---
  k_gemm_nt<<<(XROWS / 16) * (G3 / 16) / 8, blk, 0, stream>>>(X, Wih, gi, XROWS, G3, HD);

  // --- GRU recurrence: 50 sequential (1024 x 384 x 128) WMMA GEMMs + gates ---
  k_zero<<<(GRUB * HD + 255) / 256, blk, 0, stream>>>(h0, GRUB * HD);
  for (int t = 0; t < SEQL; ++t) {
    const float* Hprev = (t == 0) ? h0 : (Hbuf + (size_t)(t - 1) * GRUB * HD);
    k_gemm_nt<<<(GRUB / 16) * (G3 / 16) / 8, blk, 0, stream>>>(Hprev, Whh, gh, GRUB, G3, HD);
    k_gru_gate<<<(GRUB * HD) / 256, blk, 0, stream>>>(gi, gh, bih, bhh, Hprev,
                                                      Hbuf + (size_t)t * GRUB * HD, t);
  }

  // --- select last hidden state + LN2/LN4 ---
  k_select_ht<<<(BATCH + 7) / 8, blk, 0, stream>>>(Hbuf, lengths, ln2g, ln2b, ln4g, ln4b,
                                                   ht, htc);

  // --- dual 512 x 49999 x 128 WMMA GEMM + sigmoid-weighted combine ---
  {
    const int NTN = (NR + 15) / 16;                  // 3125
    int tiles = (BATCH / 16) * NTN;                  // 100000 wave tiles
    k_scores<<<(tiles + 7) / 8, blk, 0, stream>>>(ht, htc, iv, ctab, a1, a2, out);
  }
}